// GraphComplexityModule_30528627540049
// MI455X (gfx1250) — compile-verified
//
#include <hip/hip_runtime.h>
#include <math.h>

// ---------------------------------------------------------------------------
// GraphComplexity on MI455X (gfx1250, wave32).
// Memory-bound segmented reduction: ~308 MB traffic -> ~13 us floor @23.3TB/s.
// Phase A: per-atom neighbor counts + segment counts (wave32 segmented reduce)
// Phase B: segment sum(x), sum(x^2) with register accumulation + boundary-flush
//          atomics (segments are sorted/contiguous), float4 coalesced streams,
//          global_prefetch ahead.
// Phase C: finalize; 64-feature std reduction done with V_WMMA_F32_16X16X4_F32
//          (B = ones), one wave handles 16 segments.
// ---------------------------------------------------------------------------

typedef float v2f __attribute__((ext_vector_type(2)));
typedef float v8f __attribute__((ext_vector_type(8)));

#define FEA 64
#define MAX_NBR 12

__global__ void gc_zero_ws(float* __restrict__ p, int n) {
    int i = blockIdx.x * blockDim.x + threadIdx.x;
    if (i < n) p[i] = 0.0f;
}

// ---- Phase A: per-segment atom count + valid-neighbor count -----------------
// Lane-per-atom coalesced int4 loads; wave32 segmented suffix-reduction so only
// run-head lanes issue global atomics (~95K atomics total instead of 2M).
__global__ void gc_phaseA(const int* __restrict__ nbr,
                          const int* __restrict__ seg,
                          float* __restrict__ counts,
                          float* __restrict__ conn,
                          int N) {
    int a = blockIdx.x * blockDim.x + threadIdx.x;
    int lane = threadIdx.x & 31;

    int s = -2;            // sentinel segment for out-of-range lanes
    float v = 0.0f;        // valid-neighbor count for this atom
    float c = 0.0f;        // atom count contribution
    if (a < N) {
        s = seg[a];
        const int4* row = reinterpret_cast<const int4*>(nbr + (size_t)a * MAX_NBR);
        int4 r0 = row[0];
        int4 r1 = row[1];
        int4 r2 = row[2];
        int vc = (r0.x >= 0) + (r0.y >= 0) + (r0.z >= 0) + (r0.w >= 0)
               + (r1.x >= 0) + (r1.y >= 0) + (r1.z >= 0) + (r1.w >= 0)
               + (r2.x >= 0) + (r2.y >= 0) + (r2.z >= 0) + (r2.w >= 0);
        v = (float)vc;
        c = 1.0f;
    }

    // Segmented (run-aware) suffix reduction across the wave32.
    #pragma unroll
    for (int k = 1; k < 32; k <<= 1) {
        int   so = __shfl_down(s, k, 32);
        float vt = __shfl_down(v, k, 32);
        float ct = __shfl_down(c, k, 32);
        if ((lane + k) < 32 && so == s) { v += vt; c += ct; }
    }
    int sprev = __shfl_up(s, 1, 32);
    bool head = (lane == 0) || (sprev != s);
    if (head && s >= 0) {
        atomicAdd(&counts[s], c);
        atomicAdd(&conn[s],  v);
    }
}

// ---- Phase B: segment sum(x) and sum(x*x) ----------------------------------
// 256 threads = 16 slots x 16 lanes; each slot owns one atom per step, lane q
// owns features [4q,4q+3] via a float4 load -> each wave reads 512 contiguous
// bytes per instruction. Sorted segments => accumulate in registers, flush to
// global atomics only when the segment id changes.
__global__ void gc_phaseB(const float* __restrict__ fea,
                          const int* __restrict__ seg,
                          float* __restrict__ sumx,
                          float* __restrict__ sumx2,
                          int N) {
    const int tid  = threadIdx.x;
    const int q    = tid & 15;   // feature quad index
    const int slot = tid >> 4;   // atom slot within iteration
    const int base = blockIdx.x * 256;

    float ax = 0.f, ay = 0.f, az = 0.f, aw = 0.f;
    float bx = 0.f, by = 0.f, bz = 0.f, bw = 0.f;
    int cur = -1;

    for (int i = 0; i < 16; ++i) {
        int a = base + i * 16 + slot;
        if (a >= N) break;

        // Prefetch ~4 iterations (64 atoms) ahead: emits global_prefetch_b8.
        if (a + 64 < N)
            __builtin_prefetch(fea + (size_t)(a + 64) * FEA + q * 4, 0, 1);

        int s = seg[a];  // 16 lanes share this address -> broadcast
        const float4* row = reinterpret_cast<const float4*>(fea + (size_t)a * FEA);
        float4 x = row[q];

        if (s != cur) {
            if (cur >= 0) {
                float* px = sumx  + (size_t)cur * FEA + q * 4;
                float* p2 = sumx2 + (size_t)cur * FEA + q * 4;
                atomicAdd(px + 0, ax); atomicAdd(px + 1, ay);
                atomicAdd(px + 2, az); atomicAdd(px + 3, aw);
                atomicAdd(p2 + 0, bx); atomicAdd(p2 + 1, by);
                atomicAdd(p2 + 2, bz); atomicAdd(p2 + 3, bw);
                ax = ay = az = aw = 0.f;
                bx = by = bz = bw = 0.f;
            }
            cur = s;
        }
        ax += x.x; ay += x.y; az += x.z; aw += x.w;
        bx += x.x * x.x; by += x.y * x.y; bz += x.z * x.z; bw += x.w * x.w;
    }
    if (cur >= 0) {
        float* px = sumx  + (size_t)cur * FEA + q * 4;
        float* p2 = sumx2 + (size_t)cur * FEA + q * 4;
        atomicAdd(px + 0, ax); atomicAdd(px + 1, ay);
        atomicAdd(px + 2, az); atomicAdd(px + 3, aw);
        atomicAdd(p2 + 0, bx); atomicAdd(p2 + 1, by);
        atomicAdd(p2 + 2, bz); atomicAdd(p2 + 3, bw);
    }
}

// ---- Phase C: finalize with WMMA --------------------------------------------
// One wave handles 16 segments. A-matrix (16x4 f32, ISA layout: lanes 0-15 hold
// {K=0,K=1} for M=lane, lanes 16-31 hold {K=2,K=3} for M=lane-16) carries
// per-feature std values; B = ones, so D[m][*] = sum_f std[m][f] after 16
// chained WMMAs covering all 64 features. Lanes 0 and 16 (N==0 columns of the
// C/D layout) extract row sums and write the fused complexity.
__global__ void gc_phaseC(const float* __restrict__ counts,
                          const float* __restrict__ conn,
                          const float* __restrict__ sumx,
                          const float* __restrict__ sumx2,
                          const float* __restrict__ fw,
                          float* __restrict__ out,
                          int S) {
    int tid  = blockIdx.x * blockDim.x + threadIdx.x;
    int gw   = tid >> 5;            // global wave id
    int lane = threadIdx.x & 31;
    int g0   = gw * 16;
    if (g0 >= S) return;            // uniform per wave: EXEC stays all-ones

    int m  = lane & 15;             // matrix row / segment within group
    int kh = lane >> 4;             // which K-pair this lane supplies
    int sg = g0 + m;
    if (sg >= S) sg = S - 1;        // clamp; writes are bounds-guarded below

    float cnt  = counts[sg];
    float inv  = 1.0f / fmaxf(cnt, 1.0f);

    v8f c = {0.f, 0.f, 0.f, 0.f, 0.f, 0.f, 0.f, 0.f};
    v2f B = {1.0f, 1.0f};

    for (int t = 0; t < 16; ++t) {
        int f = 4 * t + 2 * kh;     // kh=0 -> K=0,1 ; kh=1 -> K=2,3
        size_t off = (size_t)sg * FEA + f;
        float sx0 = sumx[off],  sx1 = sumx[off + 1];
        float s20 = sumx2[off], s21 = sumx2[off + 1];
        float m0 = sx0 * inv, m1 = sx1 * inv;
        float a0 = sqrtf(fmaxf(s20 * inv - m0 * m0, 0.0f));
        float a1 = sqrtf(fmaxf(s21 * inv - m1 * m1, 0.0f));
        v2f A = {a0, a1};
        c = __builtin_amdgcn_wmma_f32_16x16x4_f32(
                /*neg_a=*/false, A, /*neg_b=*/false, B,
                /*c_mod=*/(short)0, c, /*reuse_a=*/false, /*reuse_b=*/false);
    }

    if ((lane & 15) == 0) {         // lanes 0 (M=0..7) and 16 (M=8..15)
        float w0 = fw[0], w1 = fw[1], w2 = fw[2];
        float mx = fmaxf(fmaxf(w0, w1), w2);
        float e0 = expf(w0 - mx), e1 = expf(w1 - mx), e2 = expf(w2 - mx);
        float es = e0 + e1 + e2;
        w0 = e0 / es; w1 = e1 / es; w2 = e2 / es;

        const float invL1P = 1.0f / log1pf(100.0f);  // MAX_ATOMS = 100
        int half = lane >> 4;
        #pragma unroll
        for (int r = 0; r < 8; ++r) {
            int g = g0 + half * 8 + r;
            if (g >= S) break;
            float cn = counts[g];
            float sf = fmaxf(cn, 1.0f);
            float std_mean = c[r] * (1.0f / 64.0f);
            float chem  = (cn > 1.0f)
                        ? (1.0f / (1.0f + expf(-(std_mean - 0.5f)))) : 0.0f;
            float scale = log1pf(cn) * invL1P;
            float conn_c = fminf((conn[g] / sf) * (1.0f / 12.0f), 1.0f);
            float comp = w0 * scale + w1 * chem + w2 * conn_c;
            comp = fminf(fmaxf(comp, 0.0f), 1.0f);
            out[g] = (cn > 0.0f) ? comp : 0.0f;
        }
    }
}

extern "C" void kernel_launch(void* const* d_in, const int* in_sizes, int n_in,
                              void* d_out, int out_size, void* d_ws, size_t ws_size,
                              hipStream_t stream) {
    const float* atom_fea = (const float*)d_in[0];
    const int*   nbr_idx  = (const int*)d_in[1];
    const int*   seg_ids  = (const int*)d_in[2];
    const float* fweights = (const float*)d_in[3];
    float*       out      = (float*)d_out;

    const int N = in_sizes[0] / FEA;   // atoms
    const int S = out_size;            // segments (num_segments)

    // Workspace layout (floats): counts[S] | conn[S] | sumx[S*64] | sumx2[S*64]
    float* wsf    = (float*)d_ws;
    float* counts = wsf;
    float* conn   = wsf + (size_t)S;
    float* sumx   = wsf + (size_t)2 * S;
    float* sumx2  = sumx + (size_t)S * FEA;

    const int zn = S * (2 + 2 * FEA);  // 130*S floats to zero
    gc_zero_ws<<<(zn + 255) / 256, 256, 0, stream>>>(wsf, zn);

    gc_phaseA<<<(N + 255) / 256, 256, 0, stream>>>(nbr_idx, seg_ids, counts, conn, N);
    gc_phaseB<<<(N + 255) / 256, 256, 0, stream>>>(atom_fea, seg_ids, sumx, sumx2, N);

    int waves  = (S + 15) / 16;
    int blocks = (waves * 32 + 255) / 256;
    gc_phaseC<<<blocks, 256, 0, stream>>>(counts, conn, sumx, sumx2, fweights, out, S);
}